// CNNDSTv2_batch_20796231647897
// MI455X (gfx1250) — compile-verified
//
#include <hip/hip_runtime.h>
#include <cstdint>

// CDNA5 / gfx1250: wave32, WGP with 320 KB LDS, WMMA matrix ops, async global<->LDS.
typedef __attribute__((ext_vector_type(2))) float v2f;
typedef __attribute__((ext_vector_type(8))) float v8f;

#define N_BITS 16
#define LSIZE  (1 << N_BITS)   // 65536 elements = 256 KB f32, fits one WGP's LDS
#define BATCH  512
#define TPB    1024            // 32 waves of 32

// ---------------- async global <-> LDS (ASYNCcnt path) ----------------

__device__ __forceinline__ void async_wait0() {
#if __has_builtin(__builtin_amdgcn_s_wait_asynccnt)
  __builtin_amdgcn_s_wait_asynccnt(0);
#else
  asm volatile("s_wait_asynccnt 0" ::: "memory");
#endif
}

// 256 KB global -> LDS, 16 x b128 per thread, coalesced + conflict-free.
__device__ __forceinline__ void async_copy_in(float* lds, const float* __restrict__ src, int tid) {
  const float4* g = (const float4*)src;
  unsigned ldsb = (unsigned)(uintptr_t)lds;   // generic LDS ptr: addr[31:0] is the LDS byte address
#pragma unroll
  for (int j = 0; j < 16; ++j) {
    int i4 = tid + (j << 10);
    unsigned loff = ldsb + ((unsigned)i4 << 4);
    const float4* gp = g + i4;
    asm volatile("global_load_async_to_lds_b128 %0, %1, off"
                 :: "v"(loff), "v"(gp) : "memory");
  }
  async_wait0();
  __syncthreads();
}

// 256 KB LDS -> global, straight out of LDS (no VGPR round-trip).
__device__ __forceinline__ void async_copy_out(float* __restrict__ dst, const float* lds, int tid) {
  float4* g = (float4*)dst;
  unsigned ldsb = (unsigned)(uintptr_t)lds;
#pragma unroll
  for (int j = 0; j < 16; ++j) {
    int i4 = tid + (j << 10);
    unsigned loff = ldsb + ((unsigned)i4 << 4);
    float4* gp = g + i4;
    asm volatile("global_store_async_from_lds_b128 %0, %1, off"
                 :: "v"(gp), "v"(loff) : "memory");
  }
  async_wait0();
}

// ---------------- butterfly levels 4..15 (LDS, conflict-free) ----------------

template <int SIGN>
__device__ __forceinline__ void butterfly_high(float* lds, int tid) {
#pragma unroll
  for (int k = 4; k < N_BITS; ++k) {
    const int s = 1 << k;
#pragma unroll 4
    for (int j = 0; j < 32; ++j) {            // 32768 active pairs / 1024 threads
      int idx = tid + (j << 10);
      int i = ((idx >> k) << (k + 1)) | (idx & (s - 1));  // insert 0 at bit k
      float a = lds[i];
      float c = lds[i + s];
      lds[i] = (SIGN > 0) ? (a + c) : (a - c);
    }
    __syncthreads();
  }
}

// ---------------- levels 0..3 as one WMMA pass ----------------
// Low-4-bit zeta/Mobius is y = T.x with T[j][m] = [m superset of j] (signed for Mobius).
// Per 16x16 tile (rows = bits 4..7, cols = bits 0..3): Y = X * T^T via 4 chained
// V_WMMA_F32_16X16X4_F32 (K=4 chunks). B (=T^T chunk) is a per-lane constant.
__device__ __forceinline__ void low4_wmma(float* lds, int tid, const v2f* B) {
  const int lane = tid & 31;
  const int wave = tid >> 5;
  const int hi   = lane >> 4;        // half-wave select
  const int row  = lane & 15;
#pragma unroll
  for (int j = 0; j < 8; ++j) {      // 256 tiles / 32 waves
    float* tb = lds + ((wave + (j << 5)) << 8);
    // f32 A layout 16x4: lanes 0-15 hold K = 4c+{0,1}, lanes 16-31 hold K = 4c+{2,3}
    v2f a0 = *(const v2f*)(tb + row * 16 + 0  + 2 * hi);
    v2f a1 = *(const v2f*)(tb + row * 16 + 4  + 2 * hi);
    v2f a2 = *(const v2f*)(tb + row * 16 + 8  + 2 * hi);
    v2f a3 = *(const v2f*)(tb + row * 16 + 12 + 2 * hi);
    v8f acc = {};
    acc = __builtin_amdgcn_wmma_f32_16x16x4_f32(false, a0, false, B[0], (short)0, acc, false, false);
    acc = __builtin_amdgcn_wmma_f32_16x16x4_f32(false, a1, false, B[1], (short)0, acc, false, false);
    acc = __builtin_amdgcn_wmma_f32_16x16x4_f32(false, a2, false, B[2], (short)0, acc, false, false);
    acc = __builtin_amdgcn_wmma_f32_16x16x4_f32(false, a3, false, B[3], (short)0, acc, false, false);
    // D layout: VGPR r, lanes 0-15 -> M=r, lanes 16-31 -> M=r+8; N = lane&15
#pragma unroll
    for (int r = 0; r < 8; ++r)
      tb[(r + 8 * hi) * 16 + row] = acc[r];
  }
  __syncthreads();
}

// ---------------- kernel: one batch per workgroup ----------------

__global__ void __launch_bounds__(TPB)
subset_conv_kernel(const float* __restrict__ M, float* __restrict__ out) {
  extern __shared__ float lds[];               // 65536 floats = 256 KB dynamic LDS
  const int tid  = threadIdx.x;
  const int b    = blockIdx.x;
  const int lane = tid & 31;
  const int hi   = lane >> 4;
  const int ncol = lane & 15;

  // Constant B operands (T^T chunks) for zeta and Mobius, per the f32 B VGPR layout:
  // lanes 0-15: K = 4c+{0,1}; lanes 16-31: K = 4c+{2,3}; N = lane&15.
  v2f Bz[4], Bm[4];
#pragma unroll
  for (int c = 0; c < 4; ++c) {
    int m0 = 4 * c + 2 * hi;
#pragma unroll
    for (int t = 0; t < 2; ++t) {
      int m = m0 + t;
      bool sup = ((m & ncol) == ncol);                         // m superset of j
      float z  = sup ? 1.0f : 0.0f;
      float mo = sup ? (((__popc(m ^ ncol) & 1) != 0) ? -1.0f : 1.0f) : 0.0f;
      if (t == 0) { Bz[c].x = z; Bm[c].x = mo; }
      else        { Bz[c].y = z; Bm[c].y = mo; }
    }
  }

  const float* base = M + (size_t)b * 2 * LSIZE;
  float z0r[64];                                // z0 lives in VGPRs (LDS-bound: 1 WG/WGP anyway)

  // ---- channel 0: zeta ----
  async_copy_in(lds, base, tid);
  low4_wmma(lds, tid, Bz);
  butterfly_high<+1>(lds, tid);
#pragma unroll
  for (int j = 0; j < 64; ++j) z0r[j] = lds[tid + (j << 10)];
  __syncthreads();

  // ---- channel 1: zeta, then q = z0 * z1 in place ----
  async_copy_in(lds, base + LSIZE, tid);
  low4_wmma(lds, tid, Bz);
  butterfly_high<+1>(lds, tid);
#pragma unroll
  for (int j = 0; j < 64; ++j) {
    int i = tid + (j << 10);
    lds[i] *= z0r[j];
  }
  __syncthreads();

  // ---- Mobius (inverse) ----
  low4_wmma(lds, tid, Bm);
  butterfly_high<-1>(lds, tid);

  // ---- stream result out of LDS ----
  async_copy_out(out + (size_t)b * LSIZE, lds, tid);
}

// ---------------- host entry ----------------

extern "C" void kernel_launch(void* const* d_in, const int* in_sizes, int n_in,
                              void* d_out, int out_size, void* d_ws, size_t ws_size,
                              hipStream_t stream) {
  (void)in_sizes; (void)n_in; (void)d_ws; (void)ws_size; (void)out_size;
  const float* M = (const float*)d_in[0];
  float* out = (float*)d_out;
  subset_conv_kernel<<<dim3(BATCH), dim3(TPB), LSIZE * sizeof(float), stream>>>(M, out);
}